// AsymConv_48713519071682
// MI455X (gfx1250) — compile-verified
//
#include <hip/hip_runtime.h>
#include <hip/hip_bf16.h>
#include <stdint.h>

// AsymConv: location-dependent 3x3x3 conv with replicate padding.
// X: [2,1,96,96,96] f32   W: [1,1,96,96,96,27] f32   out: [2,1,96,96,96] f32
//
// Memory-bound (AI ~0.8 FLOP/B). Strategy:
//  - one thread per (x,y,z) voxel, computes BOTH batches (W streamed once)
//  - W slice staged to LDS via global_load_async_to_lds_b128 (NT hint)
//  - replicate-padded X tile staged to LDS via global_load_async_to_lds_b32
//  - s_wait_asynccnt 0 + barrier, then unrolled 27-tap dual-batch FMA

#define D   96
#define D2  (96 * 96)
#define D3  (96 * 96 * 96)

__global__ __launch_bounds__(192) void AsymConv_kernel(
    const float* __restrict__ X,   // [2][96][96][96]
    const float* __restrict__ W,   // [96][96][96][27]
    float* __restrict__ O)         // [2][96][96][96]
{
    // W slice for this block: 2 y-lines x 96 z x 27 taps (contiguous, 16B aligned)
    __shared__ __align__(16) float Ws[2 * 96 * 27];        // 20736 B
    // Replicate-padded X tile: [batch][x-1..x+1][y0-1..y0+2][z padded 0..97]
    __shared__ __align__(16) float Xs[2][3][4][98];        // 9408 B

    const int z   = threadIdx.x;            // 0..95
    const int ty  = threadIdx.y;            // 0..1
    const int tid = ty * 96 + z;            // 0..191

    const int x  = blockIdx.y;              // 0..95
    const int y0 = blockIdx.x * 2;          // 0..94

    // ---------------- stage W via async B128 (non-temporal stream) ----------
    // Region: floats [ (x*96+y0)*96*27 , +5184 ) -> 1296 x float4, 16B aligned.
    const float* wsrc = W + (size_t)((x * 96 + y0) * 96) * 27;
    const uint32_t wlds0 = (uint32_t)(uintptr_t)(&Ws[0]);
    #pragma unroll 1
    for (int i = tid; i < 1296; i += 192) {
        uint32_t la = wlds0 + (uint32_t)i * 16u;
        unsigned long long ga = (unsigned long long)(uintptr_t)(wsrc + (size_t)i * 4);
        asm volatile("global_load_async_to_lds_b128 %0, %1, off th:TH_LOAD_NT"
                     :: "v"(la), "v"(ga) : "memory");
    }

    // ---------------- stage X tile (replicate halo) via async B32 -----------
    // 2 batches * 3 x-rows * 4 y-rows * 98 z = 2352 elements.
    const uint32_t xlds0 = (uint32_t)(uintptr_t)(&Xs[0][0][0][0]);
    #pragma unroll 1
    for (int i = tid; i < 2352; i += 192) {
        int b  = i / 1176;
        int r  = i - b * 1176;
        int xi = r / 392;
        int r2 = r - xi * 392;
        int yj = r2 / 98;
        int zp = r2 - yj * 98;
        int gx = min(max(x  - 1 + xi, 0), 95);
        int gy = min(max(y0 - 1 + yj, 0), 95);
        int gz = min(max(zp - 1,      0), 95);
        uint32_t la = xlds0 + (uint32_t)i * 4u;
        unsigned long long ga = (unsigned long long)(uintptr_t)
            (X + (((size_t)b * D + gx) * D + gy) * D + gz);
        asm volatile("global_load_async_to_lds_b32 %0, %1, off"
                     :: "v"(la), "v"(ga) : "memory");
    }

    // Async writes to LDS are invisible to the compiler's memory model:
    // wait on ASYNCcnt explicitly, then workgroup barrier.
    asm volatile("s_wait_asynccnt 0" ::: "memory");
    __syncthreads();

    // ---------------- compute: 27 taps, both batches -------------------------
    const int wb = tid * 27;   // this voxel's 27 weights in Ws
    float acc0 = 0.0f, acc1 = 0.0f;
    #pragma unroll
    for (int i = 0; i < 3; ++i) {
        #pragma unroll
        for (int j = 0; j < 3; ++j) {
            #pragma unroll
            for (int l = 0; l < 3; ++l) {
                const float w = Ws[wb + (i * 3 + j) * 3 + l];
                acc0 = fmaf(Xs[0][i][ty + j][z + l], w, acc0);
                acc1 = fmaf(Xs[1][i][ty + j][z + l], w, acc1);
            }
        }
    }

    const int y = y0 + ty;
    const size_t o = ((size_t)x * D + y) * D + z;
    O[o]      = acc0;   // batch 0
    O[o + D3] = acc1;   // batch 1
}

extern "C" void kernel_launch(void* const* d_in, const int* in_sizes, int n_in,
                              void* d_out, int out_size, void* d_ws, size_t ws_size,
                              hipStream_t stream) {
    (void)in_sizes; (void)n_in; (void)out_size; (void)d_ws; (void)ws_size;
    const float* X = (const float*)d_in[0];
    const float* W = (const float*)d_in[1];
    float*       O = (float*)d_out;

    dim3 block(96, 2, 1);     // 192 threads = 6 wave32
    dim3 grid(48, 96, 1);     // 48 y-tiles x 96 x-planes
    hipLaunchKernelGGL(AsymConv_kernel, grid, block, 0, stream, X, W, O);
}